// Filtration_31705448579348
// MI455X (gfx1250) — compile-verified
//
#include <hip/hip_runtime.h>
#include <math.h>

// ---------------------------------------------------------------------------
// GIN filtration model for MI455X (gfx1250), wave32, WMMA f32 16x16x4 path.
// DIM = 64 everywhere; L = 3.
//
// v2: - weights staged in LDS pre-swizzled as float2{W[k][n], W[k+1][n]} so
//       each WMMA B-fragment is ONE aligned ds_load_b64 (no reg shuffling),
//       bank-conflict-free across the wave (bank = (4n+2half) mod 64).
//     - A-tiles padded to stride 68 floats: bank = (4m+kk) mod 64, removing
//       the 16-way conflict of the stride-64 layout.
//     - global_prefetch_b8 on the edge stream in the scatter kernel.
// ---------------------------------------------------------------------------

typedef __attribute__((ext_vector_type(2))) float v2f;
typedef __attribute__((ext_vector_type(8))) float v8f;

#define DIM 64
#define TSTRIDE 68   // padded A-tile row stride (floats); even => 8B aligned
#define NEG_SLOPE 0.01f
#define BN_EPS 1e-5f

// --------------------------- small utility kernels -------------------------

__global__ void zero_kernel(float* __restrict__ p, size_t count) {
    size_t i = (size_t)blockIdx.x * blockDim.x + threadIdx.x;
    size_t stride = (size_t)gridDim.x * blockDim.x;
    for (; i < count; i += stride) p[i] = 0.0f;
}

// h0[i,:] = embed_deg[node_deg[i], :]   (float4-vectorized: 16 float4 per row)
__global__ void gather_embed_kernel(const int* __restrict__ node_deg,
                                    const float* __restrict__ embed,
                                    float* __restrict__ z0, int n) {
    int t = blockIdx.x * blockDim.x + threadIdx.x;
    int i = t >> 4;
    int q = t & 15;
    if (i >= n) return;
    int deg = node_deg[i];
    ((float4*)z0)[t] = ((const float4*)embed)[deg * 16 + q];
}

// agg[dst[e],:] += x[src[e],:]  — 16 threads per edge, 4 floats each.
__global__ void scatter_add_kernel(const float* __restrict__ x,
                                   const int* __restrict__ ei,
                                   float* __restrict__ agg, int nE) {
    int t = blockIdx.x * blockDim.x + threadIdx.x;
    int e = t >> 4;
    int q = t & 15;
    if (e >= nE) return;
    // prefetch the edge-index stream ahead (global_prefetch_b8)
    if (e + 8192 < nE) {
        __builtin_prefetch(&ei[e + 8192], 0, 0);
        __builtin_prefetch(&ei[nE + e + 8192], 0, 0);
    }
    int s = ei[e];
    int d = ei[nE + e];
    float4 v = ((const float4*)x)[(size_t)s * 16 + q];
    float* p = agg + (size_t)d * DIM + q * 4;
    atomicAdd(p + 0, v.x);
    atomicAdd(p + 1, v.y);
    atomicAdd(p + 2, v.z);
    atomicAdd(p + 3, v.w);
}

// ------------- packed-weight staging: one ds_load_b64 per B frag -----------
// Layout: pk[(k0/4)*128 + n*2 + half] = { W[k0+2*half][n], W[k0+2*half+1][n] }
__device__ __forceinline__ void stage_weights_packed(
    v2f* __restrict__ pk, const float* __restrict__ W, int tid, int nthreads) {
    for (int idx = tid; idx < 2048; idx += nthreads) {
        int kq = idx >> 7;          // k0/4
        int rem = idx & 127;
        int nn = rem >> 1;
        int half = rem & 1;
        int k = kq * 4 + half * 2;
        v2f w;
        w.x = W[k * DIM + nn];
        w.y = W[k * DIM + nn + DIM];
        pk[idx] = w;
    }
}

// --------------------------- fused GIN layer (WMMA) ------------------------
// Per wave: 16 rows x 64 cols. m = (1+eps)*x + agg (staged to LDS),
// h1 = leaky(m @ W1 + b1) (back to LDS), pre = h1 @ W2 + b2 -> global.
// All GEMMs via V_WMMA_F32_16X16X4_F32; K-loop of 16 per 16x16 tile.

__global__ __launch_bounds__(128) void gin_layer_kernel(
    const float* __restrict__ x, const float* __restrict__ agg,
    const float* __restrict__ W1, const float* __restrict__ b1,
    const float* __restrict__ W2, const float* __restrict__ b2,
    const float* __restrict__ eps_l, float* __restrict__ pre, int n) {
    __shared__ v2f sW1p[2048];            // 16KB, packed
    __shared__ v2f sW2p[2048];            // 16KB, packed
    __shared__ float sB1[DIM];
    __shared__ float sB2[DIM];
    __shared__ float sTile[4][16 * TSTRIDE];  // padded 16x64 tile per wave

    const int tid = threadIdx.x;
    stage_weights_packed(sW1p, W1, tid, 128);
    stage_weights_packed(sW2p, W2, tid, 128);
    if (tid < DIM) { sB1[tid] = b1[tid]; sB2[tid] = b2[tid]; }
    __syncthreads();

    const int wave = tid >> 5;
    const int lane = tid & 31;
    const int half = (lane >= 16) ? 1 : 0;  // which K-pair this lane holds
    const int mrow = lane & 15;
    const int rbase = half * 8;             // C/D: vgpr r -> row r + 8*half
    const int rowBase = blockIdx.x * 64 + wave * 16;
    const float epv = 1.0f + eps_l[0];

    float* tile = sTile[wave];

    // Build m = (1+eps)*x + agg into this wave's LDS tile (2 cols per lane).
    {
        const int c = lane * 2;
        for (int r = 0; r < 16; ++r) {
            int row = rowBase + r;
            if (row >= n) row = n - 1;  // clamp; tail rows never stored
            float2 xv = *(const float2*)(x + (size_t)row * DIM + c);
            float2 av = *(const float2*)(agg + (size_t)row * DIM + c);
            v2f m;
            m.x = epv * xv.x + av.x;
            m.y = epv * xv.y + av.y;
            *(v2f*)(tile + r * TSTRIDE + c) = m;
        }
    }

    // ---- GEMM1: h1 = leaky(m @ W1 + b1) ----
    v8f acc[4] = {};
#pragma unroll
    for (int k0 = 0; k0 < DIM; k0 += 4) {
        const int kk = k0 + half * 2;
        const v2f a = *(const v2f*)(tile + mrow * TSTRIDE + kk);
        const int pbase = (k0 >> 2) * 128 + half;
#pragma unroll
        for (int nt = 0; nt < 4; ++nt) {
            const int nn = nt * 16 + mrow;
            const v2f b = sW1p[pbase + nn * 2];
            acc[nt] = __builtin_amdgcn_wmma_f32_16x16x4_f32(
                false, a, false, b, (short)0, acc[nt], false, false);
        }
    }
    // bias + leaky, write h1 back into the same LDS tile (wave-private).
#pragma unroll
    for (int nt = 0; nt < 4; ++nt) {
        const int nn = nt * 16 + mrow;
        const float bb = sB1[nn];
#pragma unroll
        for (int r = 0; r < 8; ++r) {
            float v = acc[nt][r] + bb;
            v = (v >= 0.0f) ? v : NEG_SLOPE * v;
            tile[(rbase + r) * TSTRIDE + nn] = v;
        }
    }

    // ---- GEMM2: pre = h1 @ W2 + b2 ----
    v8f acc2[4] = {};
#pragma unroll
    for (int k0 = 0; k0 < DIM; k0 += 4) {
        const int kk = k0 + half * 2;
        const v2f a = *(const v2f*)(tile + mrow * TSTRIDE + kk);
        const int pbase = (k0 >> 2) * 128 + half;
#pragma unroll
        for (int nt = 0; nt < 4; ++nt) {
            const int nn = nt * 16 + mrow;
            const v2f b = sW2p[pbase + nn * 2];
            acc2[nt] = __builtin_amdgcn_wmma_f32_16x16x4_f32(
                false, a, false, b, (short)0, acc2[nt], false, false);
        }
    }
#pragma unroll
    for (int nt = 0; nt < 4; ++nt) {
        const int nn = nt * 16 + mrow;
        const float bb = sB2[nn];
#pragma unroll
        for (int r = 0; r < 8; ++r) {
            const int row = rowBase + rbase + r;
            if (row < n) pre[(size_t)row * DIM + nn] = acc2[nt][r] + bb;
        }
    }
}

// --------------------------- FC1 (K=256 WMMA GEMM) -------------------------
// A = concat(z0..z3) (contiguous buffers, each [n,64]); W chunked via LDS.

__global__ __launch_bounds__(128) void fc1_kernel(
    const float* __restrict__ z,      // 4 contiguous [n,64] buffers
    const float* __restrict__ fcW1,   // [256,64]
    const float* __restrict__ fcb1, float* __restrict__ pre, int n) {
    __shared__ v2f sWp[2048];
    __shared__ float sB[DIM];

    const int tid = threadIdx.x;
    if (tid < DIM) sB[tid] = fcb1[tid];
    const int wave = tid >> 5;
    const int lane = tid & 31;
    const int half = (lane >= 16) ? 1 : 0;
    const int mrow = lane & 15;
    const int rbase = half * 8;
    const int rowBase = blockIdx.x * 64 + wave * 16;
    int row = rowBase + mrow;
    if (row >= n) row = n - 1;

    v8f acc[4] = {};
    for (int kb = 0; kb < 4; ++kb) {
        __syncthreads();
        stage_weights_packed(sWp, fcW1 + kb * DIM * DIM, tid, 128);
        __syncthreads();
        const float* zb = z + (size_t)kb * (size_t)n * DIM;
#pragma unroll
        for (int k0 = 0; k0 < DIM; k0 += 4) {
            const int kk = k0 + half * 2;
            float2 av = *(const float2*)(zb + (size_t)row * DIM + kk);
            v2f a;
            a.x = av.x;
            a.y = av.y;
            const int pbase = (k0 >> 2) * 128 + half;
#pragma unroll
            for (int nt = 0; nt < 4; ++nt) {
                const int nn = nt * 16 + mrow;
                const v2f b = sWp[pbase + nn * 2];
                acc[nt] = __builtin_amdgcn_wmma_f32_16x16x4_f32(
                    false, a, false, b, (short)0, acc[nt], false, false);
            }
        }
    }
#pragma unroll
    for (int nt = 0; nt < 4; ++nt) {
        const int nn = nt * 16 + mrow;
        const float bb = sB[nn];
#pragma unroll
        for (int r = 0; r < 8; ++r) {
            const int rr = rowBase + rbase + r;
            if (rr < n) pre[(size_t)rr * DIM + nn] = acc[nt][r] + bb;
        }
    }
}

// --------------------------- batch-norm (training mode) --------------------

__global__ void bn_stats_kernel(const float* __restrict__ x,
                                float* __restrict__ stats, int n) {
    __shared__ float ssum[DIM];
    __shared__ float ssq[DIM];
    const int tid = threadIdx.x;
    if (tid < DIM) { ssum[tid] = 0.0f; ssq[tid] = 0.0f; }
    __syncthreads();
    const size_t total = (size_t)n * DIM;
    const size_t stride = (size_t)gridDim.x * blockDim.x;  // multiple of 64
    const int c = tid & 63;  // channel is invariant along the stride
    float ls = 0.0f, lq = 0.0f;
    for (size_t i = (size_t)blockIdx.x * blockDim.x + tid; i < total; i += stride) {
        float v = x[i];
        ls += v;
        lq += v * v;
    }
    atomicAdd(&ssum[c], ls);   // ds_add_f32
    atomicAdd(&ssq[c], lq);
    __syncthreads();
    if (tid < DIM) {
        atomicAdd(&stats[tid], ssum[tid]);        // global_atomic_add_f32
        atomicAdd(&stats[DIM + tid], ssq[tid]);
    }
}

__global__ void bn_apply_kernel(const float* __restrict__ pre,
                                const float* __restrict__ stats,
                                const float* __restrict__ g,
                                const float* __restrict__ b,
                                float* __restrict__ out, int n) {
    size_t i = (size_t)blockIdx.x * blockDim.x + threadIdx.x;
    if (i >= (size_t)n * DIM) return;
    const int c = (int)(i & 63);
    const float inv_n = 1.0f / (float)n;
    const float mu = stats[c] * inv_n;
    const float var = stats[DIM + c] * inv_n - mu * mu;
    float v = (pre[i] - mu) * rsqrtf(var + BN_EPS) * g[c] + b[c];
    out[i] = (v >= 0.0f) ? v : NEG_SLOPE * v;
}

// --------------------------- final FC2 + sigmoid ---------------------------

__global__ void fc2_kernel(const float* __restrict__ hfc,
                           const float* __restrict__ w2,
                           const float* __restrict__ b2,
                           float* __restrict__ out, int n) {
    int i = blockIdx.x * blockDim.x + threadIdx.x;
    if (i >= n) return;
    const float4* hv = (const float4*)(hfc + (size_t)i * DIM);
    const float4* wv = (const float4*)w2;
    float acc = 0.0f;
#pragma unroll
    for (int q = 0; q < 16; ++q) {
        float4 h = hv[q];
        float4 w = wv[q];
        acc += h.x * w.x + h.y * w.y + h.z * w.z + h.w * w.w;
    }
    acc += b2[0];
    out[i] = 1.0f / (1.0f + expf(-acc));
}

// --------------------------- host-side launcher ----------------------------

extern "C" void kernel_launch(void* const* d_in, const int* in_sizes, int n_in,
                              void* d_out, int out_size, void* d_ws,
                              size_t ws_size, hipStream_t stream) {
    const int*   node_deg = (const int*)d_in[0];
    const int*   edge_index = (const int*)d_in[1];
    const float* embed = (const float*)d_in[2];
    const float* eps = (const float*)d_in[3];
    const float* W1 = (const float*)d_in[4];
    const float* b1 = (const float*)d_in[5];
    const float* W2 = (const float*)d_in[6];
    const float* b2 = (const float*)d_in[7];
    const float* bn_g = (const float*)d_in[8];
    const float* bn_b = (const float*)d_in[9];
    const float* fcW1 = (const float*)d_in[10];
    const float* fcb1 = (const float*)d_in[11];
    const float* fc_g = (const float*)d_in[12];
    const float* fc_b = (const float*)d_in[13];
    const float* fcW2 = (const float*)d_in[14];
    const float* fcb2 = (const float*)d_in[15];
    float* out = (float*)d_out;

    const int n = in_sizes[0];
    const int nE = in_sizes[1] / 2;
    const size_t nd = (size_t)n * DIM;

    float* ws = (float*)d_ws;
    float* z     = ws;            // z0..z3 contiguous: 4*nd floats
    float* agg   = ws + 4 * nd;   // per-layer aggregate / final hfc
    float* pre   = ws + 5 * nd;   // pre-BN activations
    float* stats = ws + 6 * nd;   // 128 floats: sums then sumsqs

    // z0 = embed_deg[node_deg]
    gather_embed_kernel<<<(n * 16 + 255) / 256, 256, 0, stream>>>(node_deg, embed, z, n);

    const int gemmBlocks = (n + 63) / 64;
    for (int l = 0; l < 3; ++l) {
        float* x = z + (size_t)l * nd;
        float* znext = z + (size_t)(l + 1) * nd;
        zero_kernel<<<2048, 256, 0, stream>>>(agg, nd);
        scatter_add_kernel<<<(nE * 16 + 255) / 256, 256, 0, stream>>>(x, edge_index, agg, nE);
        gin_layer_kernel<<<gemmBlocks, 128, 0, stream>>>(
            x, agg, W1 + l * DIM * DIM, b1 + l * DIM,
            W2 + l * DIM * DIM, b2 + l * DIM, eps + l, pre, n);
        zero_kernel<<<1, 128, 0, stream>>>(stats, 2 * DIM);
        bn_stats_kernel<<<1024, 256, 0, stream>>>(pre, stats, n);
        bn_apply_kernel<<<(int)((nd + 255) / 256), 256, 0, stream>>>(
            pre, stats, bn_g + l * DIM, bn_b + l * DIM, znext, n);
    }

    // FC head: pre = xcat @ fc_W1 + fc_b1 ; hfc = leaky(BN(pre)) ; sigmoid dot
    fc1_kernel<<<gemmBlocks, 128, 0, stream>>>(z, fcW1, fcb1, pre, n);
    zero_kernel<<<1, 128, 0, stream>>>(stats, 2 * DIM);
    bn_stats_kernel<<<1024, 256, 0, stream>>>(pre, stats, n);
    bn_apply_kernel<<<(int)((nd + 255) / 256), 256, 0, stream>>>(
        pre, stats, fc_g, fc_b, agg, n);  // hfc -> agg buffer
    fc2_kernel<<<(n + 255) / 256, 256, 0, stream>>>(agg, fcW2, fcb2, out, n);
}